// ToRGB_67637144977823
// MI455X (gfx1250) — compile-verified
//
#include <hip/hip_runtime.h>

// ToRGB fused kernel for MI455X (gfx1250, wave32).
//   out[b,o,h,w] = sum_c x[b,c,h,w]*W[o,c] + conv_b[o] + bias[o] + upfirdn2d(skip)[b,o,h,w]
// Memory-bound: 268 MB of x @ 23.3 TB/s ~= 11.5 us floor. f32 WMMA 16x16x4 keeps
// exact f32 math while offloading the K=512 dot products from the VALU.

typedef float v2f __attribute__((ext_vector_type(2)));
typedef float v8f __attribute__((ext_vector_type(8)));

#define CH   512
#define HW   16384      // 128*128 output pixels per (b,o) plane
#define SHW  4096       // 64*64 skip pixels per (b,o) plane

__global__ __launch_bounds__(256) void torgb_wmma_kernel(
    const float* __restrict__ x,       // [8,512,128,128]
    const float* __restrict__ skip,    // [8,3,64,64]
    const float* __restrict__ conv_w,  // [3,512]
    const float* __restrict__ conv_b,  // [3]
    const float* __restrict__ bias,    // [1,3,1,1]
    float* __restrict__ out)           // [8,3,128,128]
{
    // Stage conv_w into LDS in WMMA A-matrix lane layout: wlds[k*16 + m],
    // rows m>=3 zero so the unused 13 rows of the 16x16 tile accumulate 0.
    __shared__ float wlds[16 * CH];    // 32 KB of the 320 KB WGP LDS
    const int t = threadIdx.x;
    for (int i = t; i < 16 * CH; i += 256) {
        const int k = i >> 4;
        const int m = i & 15;
        wlds[i] = (m < 3) ? conv_w[m * CH + k] : 0.0f;
    }
    __syncthreads();

    const int lane = t & 31;
    const int wid  = t >> 5;
    const int tile = blockIdx.x * 8 + wid;     // 8192 tiles = 8 batches * 1024
    const int b    = tile >> 10;               // batch
    const int p0   = (tile & 1023) << 4;       // first pixel of tile in plane

    const int m  = lane & 15;                  // M row (A) / N col (B,C,D)
    const int hi = lane >> 4;                  // lane-half selects K pair

    // Per-lane base pointer into x for this wave's 16-pixel column block.
    const float* xb = x + (size_t)b * CH * HW + p0 + m;

    v8f acc = {};
    #pragma unroll 8
    for (int kk = 0; kk < CH; kk += 4) {
        const int ka = kk + 2 * hi;            // K index for this lane half
        v2f a, bm;
        // A: 16x4 f32 layout — VGPR0=K(ka), VGPR1=K(ka+1), M striped on lanes.
        a.x = wlds[(ka    ) * 16 + m];
        a.y = wlds[(ka + 1) * 16 + m];
        // B: 4x16 f32 — K rows mirrored across lane halves, N on lanes 0-15.
        bm.x = xb[(size_t)(ka    ) * HW];
        bm.y = xb[(size_t)(ka + 1) * HW];
        // Prefetch the x stream 16 channels ahead into the WGP-near cache
        // (locality 3 -> WGP scope; SYS-scope prefetch stops at GL2).
        if (kk + 16 < CH) __builtin_prefetch(xb + (size_t)(ka + 16) * HW, 0, 3);
        // D = A x B + C  (16x16 f32 accumulator, 8 VGPRs)
        acc = __builtin_amdgcn_wmma_f32_16x16x4_f32(
            /*neg_a=*/false, a, /*neg_b=*/false, bm,
            /*c_mod=*/(short)0, acc, /*reuse_a=*/false, /*reuse_b=*/false);
    }

    // Epilogue: C/D layout puts (M=o, N=pixel) in VGPR o, lanes 0-15.
    if (lane < 16) {
        const int p = p0 + lane;               // pixel index in 128x128 plane
        const int h = p >> 7;
        const int w = p & 127;

        // 2x FIR upsample taps: per axis weights {1,3}/4 by parity.
        int r0, r1, c0, c1;
        float fh0, fh1, fw0, fw1;
        if (h & 1) { r0 = (h - 1) >> 1; r1 = r0 + 1; fh0 = 0.75f; fh1 = 0.25f; }
        else       { r1 = h >> 1;       r0 = r1 - 1; fh0 = 0.25f; fh1 = 0.75f; }
        if (w & 1) { c0 = (w - 1) >> 1; c1 = c0 + 1; fw0 = 0.75f; fw1 = 0.25f; }
        else       { c1 = w >> 1;       c0 = c1 - 1; fw0 = 0.25f; fw1 = 0.75f; }
        const bool vr0 = (r0 >= 0), vr1 = (r1 < 64);
        const bool vc0 = (c0 >= 0), vc1 = (c1 < 64);

        const float accv[3] = { acc[0], acc[1], acc[2] };
        #pragma unroll
        for (int o = 0; o < 3; ++o) {
            const float* sb = skip + ((size_t)b * 3 + o) * SHW;
            const float s00 = (vr0 && vc0) ? sb[r0 * 64 + c0] : 0.0f;
            const float s01 = (vr0 && vc1) ? sb[r0 * 64 + c1] : 0.0f;
            const float s10 = (vr1 && vc0) ? sb[r1 * 64 + c0] : 0.0f;
            const float s11 = (vr1 && vc1) ? sb[r1 * 64 + c1] : 0.0f;
            const float up  = fh0 * (fw0 * s00 + fw1 * s01)
                            + fh1 * (fw0 * s10 + fw1 * s11);
            const float val = accv[o] + conv_b[o] + bias[o] + up;
            out[((size_t)b * 3 + o) * HW + p] = val;
        }
    }
}

extern "C" void kernel_launch(void* const* d_in, const int* in_sizes, int n_in,
                              void* d_out, int out_size, void* d_ws, size_t ws_size,
                              hipStream_t stream) {
    const float* x      = (const float*)d_in[0];  // [8,512,128,128]
    const float* skip   = (const float*)d_in[1];  // [8,3,64,64]
    const float* conv_w = (const float*)d_in[2];  // [3,512]
    const float* conv_b = (const float*)d_in[3];  // [3]
    const float* bias   = (const float*)d_in[4];  // [1,3,1,1]
    float* out = (float*)d_out;                   // [8,3,128,128]

    // 8192 tiles (16 pixels each) -> 1024 blocks x 8 waves (wave32).
    torgb_wmma_kernel<<<dim3(1024), dim3(256), 0, stream>>>(
        x, skip, conv_w, conv_b, bias, out);
}